// PoseMamba_61727269978196
// MI455X (gfx1250) — compile-verified
//
#include <hip/hip_runtime.h>
#include <math.h>

// ---------------- model constants ----------------
#define BATCH     16
#define SEQ       1024
#define M_TOT     (BATCH * SEQ)      // 16384 rows
#define IN_DIM    768
#define DM        128                 // D_MODEL
#define DST       64                  // D_STATE
#define DCONV     4
#define NH        4
#define HD        64                  // HEADDIM
#define DI        256                 // D_INNER
#define CDIM      384                 // CONV_DIM
#define NPROJ     644                 // D_IN_PROJ (640 gemm'd + 4 dt cols)
#define EPSV      1e-5f

#define QCH       64                  // chunk length for SSD scan
#define NCH       (SEQ / QCH)         // 16 chunks

// epilogue flags for the generic WMMA GEMM
#define ACT_NONE   0
#define ACT_LRELU  1
#define ACT_POSENC 2

typedef __attribute__((ext_vector_type(2))) float v2f;
typedef __attribute__((ext_vector_type(8))) float v8f;

// =====================================================================
// Generic fp32 WMMA GEMM:  out[M,N] = A[M,K] @ W[K,N] (+bias)(+addend)(+act)
// V_WMMA_F32_16X16X4_F32, wave32; one 16x16 C tile per wave.
// =====================================================================
__global__ __launch_bounds__(128)
void pm_gemm_wmma(const float* __restrict__ A, int lda,
                  const float* __restrict__ W, int ldw,
                  const float* __restrict__ bias,
                  const float* __restrict__ addend,
                  float* __restrict__ out, int ldo,
                  int M, int N, int K, int act)
{
    extern __shared__ float lds[];            // 16 * (K+1) floats
    const int m0   = blockIdx.x * 16;
    const int wave = threadIdx.x >> 5;
    const int lane = threadIdx.x & 31;
    const int hf   = lane >> 4;
    const int ml   = lane & 15;
    const int n0   = blockIdx.y * 64 + wave * 16;

    const int stride = K + 1;
    for (int idx = threadIdx.x; idx < 16 * K; idx += 128) {
        int r = idx / K, c = idx - r * K;
        lds[r * stride + c] = A[(size_t)(m0 + r) * lda + c];
    }
    __syncthreads();
    if (n0 >= N) return;

    v8f acc = {};
    for (int k = 0; k < K; k += 4) {
        int kb = k + (hf << 1);
        v2f a, b;
        a.x = lds[ml * stride + kb];
        a.y = lds[ml * stride + kb + 1];
        const float* wp = W + (size_t)kb * ldw + n0 + ml;
        b.x = wp[0];
        b.y = wp[ldw];
        acc = __builtin_amdgcn_wmma_f32_16x16x4_f32(
                  false, a, false, b, (short)0, acc, false, false);
    }

    const int col = n0 + ml;
    #pragma unroll
    for (int r = 0; r < 8; ++r) {
        int row = m0 + r + hf * 8;
        float v = acc[r];
        if (bias)   v += bias[col];
        if (addend) v += addend[(size_t)row * ldo + col];
        if (act == ACT_LRELU) v = (v > 0.f) ? v : 0.1f * v;
        else if (act == ACT_POSENC) {
            int l = row & (SEQ - 1);
            float ang = (float)l * __expf(-(float)(col & ~1) * (9.210340372f / (float)DM));
            v += (col & 1) ? __cosf(ang) : __sinf(ang);
        }
        out[(size_t)row * ldo + col] = v;
    }
}

// =====================================================================
// LayerNorm over last dim (128), one block per row
// =====================================================================
__global__ __launch_bounds__(128)
void pm_layernorm(const float* __restrict__ h,
                  const float* __restrict__ w, const float* __restrict__ b,
                  float* __restrict__ hn)
{
    __shared__ float red[DM];
    const int m = blockIdx.x, c = threadIdx.x;
    float v = h[(size_t)m * DM + c];
    red[c] = v; __syncthreads();
    for (int s = 64; s > 0; s >>= 1) { if (c < s) red[c] += red[c + s]; __syncthreads(); }
    float mu = red[0] * (1.0f / DM);
    __syncthreads();
    float d = v - mu;
    red[c] = d * d; __syncthreads();
    for (int s = 64; s > 0; s >>= 1) { if (c < s) red[c] += red[c + s]; __syncthreads(); }
    float var = red[0] * (1.0f / DM);
    hn[(size_t)m * DM + c] = d * rsqrtf(var + EPSV) * w[c] + b[c];
}

// =====================================================================
// dt = softplus(hn @ in_W[:,640:644] + dt_bias)
// =====================================================================
__global__ void pm_dt(const float* __restrict__ hn, const float* __restrict__ inW,
                      const float* __restrict__ dt_bias, float* __restrict__ dt)
{
    int m = blockIdx.x * blockDim.x + threadIdx.x;
    if (m >= M_TOT) return;
    const float* row = hn + (size_t)m * DM;
    #pragma unroll
    for (int h = 0; h < NH; ++h) {
        float acc = dt_bias[h];
        for (int k = 0; k < DM; ++k) acc += row[k] * inW[(size_t)k * NPROJ + 640 + h];
        dt[(size_t)m * NH + h] = (acc > 20.f) ? acc : log1pf(__expf(acc));
    }
}

// =====================================================================
// depthwise causal conv (4 taps) + SiLU ; xBC = zxbc[:,256:640]
// =====================================================================
__global__ void pm_conv(const float* __restrict__ zxbc,
                        const float* __restrict__ cw, const float* __restrict__ cb,
                        float* __restrict__ xc)
{
    int gid = blockIdx.x * blockDim.x + threadIdx.x;
    if (gid >= M_TOT * CDIM) return;
    int m = gid / CDIM, c = gid - m * CDIM;
    int b = m >> 10, l = m & (SEQ - 1);
    float acc = cb[c];
    #pragma unroll
    for (int k = 0; k < DCONV; ++k) {
        int ll = l + k - (DCONV - 1);
        if (ll >= 0)
            acc += cw[k * CDIM + c] * zxbc[(size_t)((b << 10) + ll) * 640 + 256 + c];
    }
    xc[(size_t)m * CDIM + c] = acc * (1.0f / (1.0f + __expf(-acc)));
}

// =====================================================================
// SSD scan stage 1: per-chunk inclusive prefix sum of a_t = dt_t * A
// grid = B*NH*NCH blocks, 64 threads
// =====================================================================
__global__ __launch_bounds__(64)
void pm_ssd_cumsum(const float* __restrict__ dt, const float* __restrict__ A_log,
                   float* __restrict__ cum)
{
    __shared__ float sa[QCH];
    const int bhc = blockIdx.x;
    const int c  = bhc & (NCH - 1);
    const int bh = bhc >> 4;
    const int hh = bh & 3;
    const int b  = bh >> 2;
    const int t  = threadIdx.x;
    const int m  = (b << 10) + c * QCH + t;

    const float Aval = -__expf(A_log[hh]);
    sa[t] = dt[(size_t)m * NH + hh] * Aval;
    __syncthreads();
    for (int off = 1; off < QCH; off <<= 1) {
        float v = (t >= off) ? sa[t - off] : 0.f;
        __syncthreads();
        sa[t] += v;
        __syncthreads();
    }
    cum[(size_t)bhc * QCH + t] = sa[t];
}

// =====================================================================
// SSD stage 2: chunk state contribution (WMMA GEMM 64x64x64)
//   G[p,n] = sum_s (dt_s * exp(csum - cum[s]) * x_s[p]) * B_s[n]
// grid = B*NH*NCH blocks, 128 threads (4 waves; wave = 16-col n strip)
// =====================================================================
__global__ __launch_bounds__(128)
void pm_ssd_G(const float* __restrict__ xc, const float* __restrict__ dt,
              const float* __restrict__ cum, float* __restrict__ G)
{
    __shared__ float Xl[QCH][HD + 1];    // x[s][p]
    __shared__ float Bl[QCH][DST + 1];   // B[s][n]
    __shared__ float cuml[QCH], coef[QCH];

    const int bhc = blockIdx.x;
    const int c  = bhc & (NCH - 1);
    const int bh = bhc >> 4;
    const int hh = bh & 3;
    const int b  = bh >> 2;
    const int tid  = threadIdx.x;
    const int wave = tid >> 5;
    const int lane = tid & 31;
    const int hf   = lane >> 4;
    const int ml   = lane & 15;
    const int m0g  = (b << 10) + c * QCH;

    const float* xrow = xc + (size_t)m0g * CDIM;
    for (int idx = tid; idx < QCH * 64; idx += 128) {
        int r = idx >> 6, cc = idx & 63;
        Xl[r][cc] = xrow[(size_t)r * CDIM + hh * HD + cc];
        Bl[r][cc] = xrow[(size_t)r * CDIM + DI + cc];
    }
    if (tid < QCH) cuml[tid] = cum[(size_t)bhc * QCH + tid];
    __syncthreads();
    if (tid < QCH)
        coef[tid] = dt[(size_t)(m0g + tid) * NH + hh] * __expf(cuml[QCH - 1] - cuml[tid]);
    __syncthreads();

    const int n0 = wave * 16;
    #pragma unroll
    for (int mt = 0; mt < 4; ++mt) {              // 4 row tiles of p
        v8f acc = {};
        const int prow = mt * 16 + ml;
        for (int k = 0; k < QCH; k += 4) {
            int kb = k + (hf << 1);
            v2f a, bf;
            a.x  = coef[kb]     * Xl[kb][prow];
            a.y  = coef[kb + 1] * Xl[kb + 1][prow];
            bf.x = Bl[kb][n0 + ml];
            bf.y = Bl[kb + 1][n0 + ml];
            acc = __builtin_amdgcn_wmma_f32_16x16x4_f32(
                      false, a, false, bf, (short)0, acc, false, false);
        }
        float* gp = G + (size_t)bhc * (HD * DST);
        #pragma unroll
        for (int r = 0; r < 8; ++r) {
            int p = mt * 16 + r + hf * 8;
            gp[p * DST + n0 + ml] = acc[r];
        }
    }
}

// =====================================================================
// SSD stage 3: 16-step chunk recurrence  h <- exp(csum)*h + G
// Writes h at chunk START (hprev) for every chunk.
// grid = B*NH blocks, 256 threads; each thread owns 16 state elems.
// =====================================================================
__global__ __launch_bounds__(256)
void pm_ssd_state(const float* __restrict__ G, const float* __restrict__ cum,
                  float* __restrict__ hprev)
{
    const int bh  = blockIdx.x;
    const int tid = threadIdx.x;
    float hreg[16];
    #pragma unroll
    for (int j = 0; j < 16; ++j) hreg[j] = 0.f;

    for (int c = 0; c < NCH; ++c) {
        const int bhc = bh * NCH + c;
        const size_t base = (size_t)bhc * (HD * DST) + (size_t)tid * 16;
        #pragma unroll
        for (int j = 0; j < 16; ++j) hprev[base + j] = hreg[j];
        float csE = __expf(cum[(size_t)bhc * QCH + QCH - 1]);
        #pragma unroll
        for (int j = 0; j < 16; ++j) hreg[j] = csE * hreg[j] + G[base + j];
    }
}

// =====================================================================
// SSD stage 4: per-chunk output (all WMMA):
//   S[t][s]  = C_t . B_s                       (GEMM, K=n)
//   M[t][s]  = (s<=t) ? exp(cum[t]-cum[s])*S : 0
//   Y[t][p]  = M @ (dt*X)  +  (exp(cum[t])*C) @ hprev^T  + Dp*x[t][p]
// grid = B*NH*NCH blocks, 128 threads (4 waves)
// =====================================================================
__global__ __launch_bounds__(128)
void pm_ssd_Y(const float* __restrict__ xc, const float* __restrict__ dt,
              const float* __restrict__ cum, const float* __restrict__ hprev,
              const float* __restrict__ Dp, float* __restrict__ y)
{
    __shared__ float Cl[QCH][DST + 1];   // C[t][n]
    __shared__ float Bl[QCH][DST + 1];   // B[s][n]  -> reused as M[t][s]
    __shared__ float Xl[QCH][HD + 1];    // x[s][p]
    __shared__ float cuml[QCH], dtl[QCH];

    const int bhc = blockIdx.x;
    const int c  = bhc & (NCH - 1);
    const int bh = bhc >> 4;
    const int hh = bh & 3;
    const int b  = bh >> 2;
    const int tid  = threadIdx.x;
    const int wave = tid >> 5;
    const int lane = tid & 31;
    const int hf   = lane >> 4;
    const int ml   = lane & 15;
    const int m0g  = (b << 10) + c * QCH;

    const float* xrow = xc + (size_t)m0g * CDIM;
    for (int idx = tid; idx < QCH * 64; idx += 128) {
        int r = idx >> 6, cc = idx & 63;
        Xl[r][cc] = xrow[(size_t)r * CDIM + hh * HD + cc];
        Bl[r][cc] = xrow[(size_t)r * CDIM + DI + cc];
        Cl[r][cc] = xrow[(size_t)r * CDIM + DI + DST + cc];
    }
    if (tid < QCH)       cuml[tid] = cum[(size_t)bhc * QCH + tid];
    else if (tid < 2*QCH) dtl[tid - QCH] = dt[(size_t)(m0g + tid - QCH) * NH + hh];
    __syncthreads();

    const int s0 = wave * 16;            // this wave's 16-wide strip (s, then p)

    // ---- GEMM1: S = C . B^T  (keep 4 row-tiles in registers) ----
    v8f accS[4];
    #pragma unroll
    for (int mt = 0; mt < 4; ++mt) {
        v8f acc = {};
        const int trow = mt * 16 + ml;
        for (int k = 0; k < DST; k += 4) {
            int kb = k + (hf << 1);
            v2f a, bf;
            a.x  = Cl[trow][kb];
            a.y  = Cl[trow][kb + 1];
            bf.x = Bl[s0 + ml][kb];      // B^T[k][s] = B[s][k]
            bf.y = Bl[s0 + ml][kb + 1];
            acc = __builtin_amdgcn_wmma_f32_16x16x4_f32(
                      false, a, false, bf, (short)0, acc, false, false);
        }
        accS[mt] = acc;
    }
    __syncthreads();

    // ---- mask + decay scale, write M over Bl ----
    #pragma unroll
    for (int mt = 0; mt < 4; ++mt) {
        #pragma unroll
        for (int r = 0; r < 8; ++r) {
            int t = mt * 16 + r + hf * 8;
            int s = s0 + ml;
            Bl[t][s] = (s <= t) ? __expf(cuml[t] - cuml[s]) * accS[mt][r] : 0.f;
        }
    }
    __syncthreads();

    // ---- GEMM2: Y = M @ (dt*X) + (e^cum * C) @ hprev^T + Dp*x ----
    const float* hg = hprev + (size_t)bhc * (HD * DST);
    const float Dpv = Dp[hh];
    #pragma unroll
    for (int mt = 0; mt < 4; ++mt) {
        v8f acc = {};
        const int trow = mt * 16 + ml;
        const float em = __expf(cuml[trow]);
        for (int k = 0; k < QCH; k += 4) {           // intra-chunk, K = s
            int kb = k + (hf << 1);
            v2f a, bf;
            a.x  = Bl[trow][kb];
            a.y  = Bl[trow][kb + 1];
            bf.x = dtl[kb]     * Xl[kb][s0 + ml];
            bf.y = dtl[kb + 1] * Xl[kb + 1][s0 + ml];
            acc = __builtin_amdgcn_wmma_f32_16x16x4_f32(
                      false, a, false, bf, (short)0, acc, false, false);
        }
        for (int k = 0; k < DST; k += 4) {           // inter-chunk, K = n
            int kb = k + (hf << 1);
            v2f a, bf;
            a.x  = em * Cl[trow][kb];
            a.y  = em * Cl[trow][kb + 1];
            bf.x = hg[(s0 + ml) * DST + kb];         // hprev^T[n][p] = hprev[p][n]
            bf.y = hg[(s0 + ml) * DST + kb + 1];
            acc = __builtin_amdgcn_wmma_f32_16x16x4_f32(
                      false, a, false, bf, (short)0, acc, false, false);
        }
        #pragma unroll
        for (int r = 0; r < 8; ++r) {
            int t = mt * 16 + r + hf * 8;
            int p = s0 + ml;
            float yv = acc[r] + Dpv * Xl[t][p];
            y[(size_t)(m0g + t) * DI + hh * HD + p] = yv;
        }
    }
}

// =====================================================================
// y = y * silu(z); RMS-normalize over 256; * rms_w  (in place)
// =====================================================================
__global__ __launch_bounds__(256)
void pm_gate_rms(float* __restrict__ y, const float* __restrict__ zxbc,
                 const float* __restrict__ rms_w)
{
    __shared__ float red[DI];
    const int m = blockIdx.x, c = threadIdx.x;
    float v  = y[(size_t)m * DI + c];
    float zc = zxbc[(size_t)m * 640 + c];
    float g  = v * (zc * (1.0f / (1.0f + __expf(-zc))));
    red[c] = g * g; __syncthreads();
    for (int s = 128; s > 0; s >>= 1) { if (c < s) red[c] += red[c + s]; __syncthreads(); }
    float rms = rsqrtf(red[0] * (1.0f / DI) + EPSV);
    y[(size_t)m * DI + c] = g * rms * rms_w[c];
}

// =====================================================================
// final tiny GEMM: out[M,6] = t[M,128] @ W2[128,6] + b2
// =====================================================================
__global__ void pm_fc2out(const float* __restrict__ t, const float* __restrict__ W2,
                          const float* __restrict__ b2, float* __restrict__ out)
{
    int gid = blockIdx.x * blockDim.x + threadIdx.x;
    if (gid >= M_TOT * 6) return;
    int m = gid / 6, j = gid - m * 6;
    const float* row = t + (size_t)m * DM;
    float acc = b2[j];
    for (int k = 0; k < DM; ++k) acc += row[k] * W2[k * 6 + j];
    out[gid] = acc;
}

// =====================================================================
extern "C" void kernel_launch(void* const* d_in, const int* in_sizes, int n_in,
                              void* d_out, int out_size, void* d_ws, size_t ws_size,
                              hipStream_t stream)
{
    const float* x       = (const float*)d_in[0];
    const float* fc1_W   = (const float*)d_in[1];
    const float* fc1_b   = (const float*)d_in[2];
    const float* ln_w    = (const float*)d_in[3];
    const float* ln_b    = (const float*)d_in[4];
    const float* in_W    = (const float*)d_in[5];
    const float* conv_w  = (const float*)d_in[6];
    const float* conv_b  = (const float*)d_in[7];
    const float* dt_bias = (const float*)d_in[8];
    const float* A_log   = (const float*)d_in[9];
    const float* Dp      = (const float*)d_in[10];
    const float* rms_w   = (const float*)d_in[11];
    const float* out_W   = (const float*)d_in[12];
    const float* fc2_W1  = (const float*)d_in[13];
    const float* fc2_b1  = (const float*)d_in[14];
    const float* fc2_W2  = (const float*)d_in[15];
    const float* fc2_b2  = (const float*)d_in[16];

    // workspace carve-out (fp32)
    const int NBHC = BATCH * NH * NCH;          // 1024 chunk-blocks
    float* ws    = (float*)d_ws;
    float* h     = ws;                     size_t o = (size_t)M_TOT * DM;
    float* hn    = ws + o;                 o += (size_t)M_TOT * DM;
    float* zxbc  = ws + o;                 o += (size_t)M_TOT * 640;
    float* dtb   = ws + o;                 o += (size_t)M_TOT * NH;
    float* xcb   = ws + o;                 o += (size_t)M_TOT * CDIM;
    float* yb    = ws + o;                 o += (size_t)M_TOT * DI;
    float* cumb  = ws + o;                 o += (size_t)NBHC * QCH;
    float* Gb    = ws + o;                 o += (size_t)NBHC * HD * DST;
    float* hprev = ws + o;                 /* o += NBHC * HD * DST */

    const int TILES_M = M_TOT / 16;        // 1024

    // fc1 + positional encoding
    pm_gemm_wmma<<<dim3(TILES_M, DM / 64), 128, 16 * (IN_DIM + 1) * sizeof(float), stream>>>(
        x, IN_DIM, fc1_W, DM, fc1_b, nullptr, h, DM, M_TOT, DM, IN_DIM, ACT_POSENC);

    for (int i = 0; i < 2; ++i) {
        const float* inWi = in_W + (size_t)i * DM * NPROJ;

        pm_layernorm<<<M_TOT, DM, 0, stream>>>(h, ln_w + i * DM, ln_b + i * DM, hn);

        pm_gemm_wmma<<<dim3(TILES_M, 640 / 64), 128, 16 * (DM + 1) * sizeof(float), stream>>>(
            hn, DM, inWi, NPROJ, nullptr, nullptr, zxbc, 640, M_TOT, 640, DM, ACT_NONE);

        pm_dt<<<(M_TOT + 255) / 256, 256, 0, stream>>>(hn, inWi, dt_bias + i * NH, dtb);

        pm_conv<<<(M_TOT * CDIM + 255) / 256, 256, 0, stream>>>(
            zxbc, conv_w + (size_t)i * DCONV * CDIM, conv_b + i * CDIM, xcb);

        // ---- chunked SSD scan (WMMA) ----
        pm_ssd_cumsum<<<NBHC, QCH, 0, stream>>>(dtb, A_log + i * NH, cumb);
        pm_ssd_G     <<<NBHC, 128, 0, stream>>>(xcb, dtb, cumb, Gb);
        pm_ssd_state <<<BATCH * NH, 256, 0, stream>>>(Gb, cumb, hprev);
        pm_ssd_Y     <<<NBHC, 128, 0, stream>>>(xcb, dtb, cumb, hprev, Dp + i * NH, yb);

        pm_gate_rms<<<M_TOT, DI, 0, stream>>>(yb, zxbc, rms_w + i * DI);

        pm_gemm_wmma<<<dim3(TILES_M, DM / 64), 128, 16 * (DI + 1) * sizeof(float), stream>>>(
            yb, DI, out_W + (size_t)i * DI * DM, DM, nullptr, h, h, DM, M_TOT, DM, DI, ACT_NONE);
    }

    pm_gemm_wmma<<<dim3(TILES_M, DM / 64), 128, 16 * (DM + 1) * sizeof(float), stream>>>(
        h, DM, fc2_W1, DM, fc2_b1, nullptr, hn, DM, M_TOT, DM, DM, ACT_LRELU);

    pm_fc2out<<<(M_TOT * 6 + 255) / 256, 256, 0, stream>>>(hn, fc2_W2, fc2_b2, (float*)d_out);
}